// RetNetBlock_644245094446
// MI455X (gfx1250) — compile-verified
//
#include <hip/hip_runtime.h>
#include <math.h>

#define D_DIM   512
#define T_DIM   4096
#define B_DIM   8
#define HID_DIM 1024
#define KCONV   5
#define NCHUNK  64
#define LCHUNK  64
#define LN_EPS  1e-5f

typedef __attribute__((ext_vector_type(2))) float v2f;
typedef __attribute__((ext_vector_type(4))) float v4f;
typedef __attribute__((ext_vector_type(8))) float v8f;
typedef __attribute__((ext_vector_type(4))) int   v4i;

// ---------------------------------------------------------------------------
// LayerNorm over last dim (D=512). One 256-thread block per row, 2 elems/thread.
// ---------------------------------------------------------------------------
__global__ __launch_bounds__(256) void ln_kernel(const float* __restrict__ in,
                                                 float* __restrict__ out,
                                                 const float* __restrict__ g,
                                                 const float* __restrict__ b) {
  __shared__ float red[8];
  const size_t row = blockIdx.x;
  const float* rp = in + row * D_DIM;
  float* op = out + row * D_DIM;
  const int tid = threadIdx.x;

  float v0 = rp[tid];
  float v1 = rp[tid + 256];
  float s = v0 + v1;
#pragma unroll
  for (int m = 16; m; m >>= 1) s += __shfl_xor(s, m, 32);
  if ((tid & 31) == 0) red[tid >> 5] = s;
  __syncthreads();
  float tot = 0.f;
#pragma unroll
  for (int i = 0; i < 8; i++) tot += red[i];
  const float mu = tot * (1.0f / D_DIM);

  const float d0 = v0 - mu, d1 = v1 - mu;
  float q = d0 * d0 + d1 * d1;
#pragma unroll
  for (int m = 16; m; m >>= 1) q += __shfl_xor(q, m, 32);
  __syncthreads();
  if ((tid & 31) == 0) red[tid >> 5] = q;
  __syncthreads();
  float vtot = 0.f;
#pragma unroll
  for (int i = 0; i < 8; i++) vtot += red[i];
  const float rstd = rsqrtf(vtot * (1.0f / D_DIM) + LN_EPS);

  op[tid]       = d0 * rstd * g[tid]       + b[tid];
  op[tid + 256] = d1 * rstd * g[tid + 256] + b[tid + 256];
}

// ---------------------------------------------------------------------------
// Chunked EMA scan: s_t = a*s_{t-1} + (1-a)*x_t  with a = sigmoid(decay_logit[d])
// ---------------------------------------------------------------------------
__global__ __launch_bounds__(512) void scan_phase1(const float* __restrict__ xn,
                                                   const float* __restrict__ dl,
                                                   float* __restrict__ ss,
                                                   float* __restrict__ carry) {
  const int b = blockIdx.x / NCHUNK;
  const int j = blockIdx.x % NCHUNK;
  const int d = threadIdx.x;
  const float a = 1.0f / (1.0f + expf(-dl[d]));
  const float oma = 1.0f - a;
  float s = 0.f;
  size_t base = ((size_t)b * T_DIM + (size_t)j * LCHUNK) * D_DIM + d;
  for (int k = 0; k < LCHUNK; k++) {
    s = a * s + oma * xn[base];
    ss[base] = s;
    base += D_DIM;
  }
  carry[((size_t)b * NCHUNK + j) * D_DIM + d] = s;
}

__global__ __launch_bounds__(512) void scan_phase2(const float* __restrict__ dl,
                                                   float* __restrict__ carry) {
  const int b = blockIdx.x;
  const int d = threadIdx.x;
  const float a = 1.0f / (1.0f + expf(-dl[d]));
  const float aL = powf(a, (float)LCHUNK);
  float S = 0.f;
  for (int j = 0; j < NCHUNK; j++) {
    const size_t idx = ((size_t)b * NCHUNK + j) * D_DIM + d;
    const float c = carry[idx];
    carry[idx] = S;  // state just before chunk j
    S = aL * S + c;
  }
}

__global__ __launch_bounds__(512) void scan_phase3(const float* __restrict__ dl,
                                                   const float* __restrict__ carry,
                                                   float* __restrict__ ss) {
  const int b = blockIdx.x / NCHUNK;
  const int j = blockIdx.x % NCHUNK;
  const int d = threadIdx.x;
  const float a = 1.0f / (1.0f + expf(-dl[d]));
  const float S = carry[((size_t)b * NCHUNK + j) * D_DIM + d];
  float p = a;
  size_t base = ((size_t)b * T_DIM + (size_t)j * LCHUNK) * D_DIM + d;
  for (int k = 0; k < LCHUNK; k++) {
    ss[base] += p * S;
    p *= a;
    base += D_DIM;
  }
}

// ---------------------------------------------------------------------------
// x1 = x + depthwise_conv1d(xn, K=5, same) + conv_b   (b_out added in GEMM1)
// ---------------------------------------------------------------------------
__global__ __launch_bounds__(256) void conv_res_kernel(const float* __restrict__ x,
                                                       const float* __restrict__ xn,
                                                       const float* __restrict__ cw,
                                                       const float* __restrict__ cb,
                                                       float* __restrict__ x1) {
  const size_t i = (size_t)blockIdx.x * 256 + threadIdx.x;
  const int d = (int)(i % D_DIM);
  const int t = (int)((i / D_DIM) % T_DIM);
  const size_t bt0 = (i / D_DIM) - (size_t)t;  // b*T
  float acc = x[i] + cb[d];
#pragma unroll
  for (int j = 0; j < KCONV; j++) {
    const int tt = t + j - 2;
    if (tt >= 0 && tt < T_DIM)
      acc += xn[(bt0 + (size_t)tt) * D_DIM + d] * cw[d * KCONV + j];
  }
  x1[i] = acc;
}

// ---------------------------------------------------------------------------
// fp32 WMMA GEMM: out[m,n] = epilogue( sum_k A[m,k]*W[n,k] + bias[n] )
// Block: 256 threads (8 wave32), tile 128x128, K-panel 16, LDS double-buffered
// via GLOBAL_LOAD_ASYNC_TO_LDS_B128 (ASYNCcnt) when available.
// Wave tile 32x64 => acc[2][4] of v8f; v_wmma_f32_16x16x4_f32 (exact fp32).
// ---------------------------------------------------------------------------
#define KC  16
#define LDT 20  // padded LDS row stride (floats): 80B => 16B-aligned, bank-spread

#if __has_builtin(__builtin_amdgcn_global_load_async_to_lds_b128)
#define USE_ASYNC_COPY 1
#else
#define USE_ASYNC_COPY 0
#endif

#if __has_builtin(__builtin_amdgcn_s_wait_asynccnt)
#define WAIT_ASYNC(n) __builtin_amdgcn_s_wait_asynccnt(n)
#else
#define WAIT_ASYNC(n) asm volatile("s_wait_asynccnt %0" ::"n"(n))
#endif

#if USE_ASYNC_COPY
__device__ __forceinline__ void async_b128(const float* g, float* l) {
  __builtin_amdgcn_global_load_async_to_lds_b128(
      (__attribute__((address_space(1))) v4i*)(g),
      (__attribute__((address_space(3))) v4i*)(l), 0, 0);
}
#endif

__device__ __forceinline__ void mma_chunk(const float* la, const float* lb,
                                          v8f acc[2][4], int wm, int wn,
                                          int lrow, int khalf) {
#pragma unroll
  for (int ks = 0; ks < KC / 4; ks++) {
    const int kb = ks * 4 + khalf * 2;  // lane holds K = {kb, kb+1}
    v2f af[2], bf[4];
#pragma unroll
    for (int i = 0; i < 2; i++)
      af[i] = *(const v2f*)(la + (wm * 32 + i * 16 + lrow) * LDT + kb);
#pragma unroll
    for (int j = 0; j < 4; j++)
      bf[j] = *(const v2f*)(lb + (wn * 64 + j * 16 + lrow) * LDT + kb);
#pragma unroll
    for (int i = 0; i < 2; i++)
#pragma unroll
      for (int j = 0; j < 4; j++)
        acc[i][j] = __builtin_amdgcn_wmma_f32_16x16x4_f32(
            false, af[i], false, bf[j], (short)0, acc[i][j], false, false);
  }
}

template <bool GELU>
__global__ __launch_bounds__(256) void gemm_f32_wmma(const float* __restrict__ A,
                                                     const float* __restrict__ W,
                                                     const float* __restrict__ bias,
                                                     const float* __restrict__ res,
                                                     float* __restrict__ out,
                                                     int M, int N, int K) {
#if USE_ASYNC_COPY
  __shared__ float lsA[2][128 * LDT];
  __shared__ float lsB[2][128 * LDT];
#else
  __shared__ float lsA[1][128 * LDT];
  __shared__ float lsB[1][128 * LDT];
#endif
  const int m0 = blockIdx.x * 128;
  const int n0 = blockIdx.y * 128;
  const int tid = threadIdx.x;
  const int lane = tid & 31;
  const int wave = tid >> 5;
  const int wm = wave >> 1;   // 0..3  (M direction)
  const int wn = wave & 1;    // 0..1  (N direction)
  const int lrow = lane & 15;
  const int khalf = lane >> 4;

  v8f acc[2][4];
#pragma unroll
  for (int i = 0; i < 2; i++)
#pragma unroll
    for (int j = 0; j < 4; j++) {
      v8f z = {0.f, 0.f, 0.f, 0.f, 0.f, 0.f, 0.f, 0.f};
      acc[i][j] = z;
    }

  const int nch = K / KC;

#if USE_ASYNC_COPY
  // Prologue: issue chunk 0 (4 async b128 per thread => full 128x16 A+B panels).
#pragma unroll
  for (int i = 0; i < 2; i++) {
    const int idx4 = tid + i * 256;      // 0..511
    const int r = idx4 >> 2;             // 0..127
    const int c = (idx4 & 3) << 2;       // 0,4,8,12
    async_b128(A + (size_t)(m0 + r) * K + c, &lsA[0][r * LDT + c]);
    async_b128(W + (size_t)(n0 + r) * K + c, &lsB[0][r * LDT + c]);
  }
  for (int ch = 0; ch < nch; ch++) {
    const int cur = ch & 1;
    if (ch + 1 < nch) {
      const int kc = (ch + 1) * KC;
#pragma unroll
      for (int i = 0; i < 2; i++) {
        const int idx4 = tid + i * 256;
        const int r = idx4 >> 2;
        const int c = (idx4 & 3) << 2;
        async_b128(A + (size_t)(m0 + r) * K + kc + c, &lsA[cur ^ 1][r * LDT + c]);
        async_b128(W + (size_t)(n0 + r) * K + kc + c, &lsB[cur ^ 1][r * LDT + c]);
      }
      WAIT_ASYNC(4);  // leave next chunk's 4 in flight; current chunk landed
    } else {
      WAIT_ASYNC(0);
    }
    __syncthreads();  // all waves' async loads for `cur` visible
    mma_chunk(lsA[cur], lsB[cur], acc, wm, wn, lrow, khalf);
    __syncthreads();  // done reading `cur` before it is overwritten
  }
#else
  for (int kc = 0; kc < K; kc += KC) {
    v4f ta[2], tb[2];
#pragma unroll
    for (int i = 0; i < 2; i++) {
      const int idx4 = tid + i * 256;
      const int r = idx4 >> 2;
      const int c = (idx4 & 3) << 2;
      ta[i] = *(const v4f*)(A + (size_t)(m0 + r) * K + kc + c);
      tb[i] = *(const v4f*)(W + (size_t)(n0 + r) * K + kc + c);
    }
    __syncthreads();  // previous chunk's readers done
#pragma unroll
    for (int i = 0; i < 2; i++) {
      const int idx4 = tid + i * 256;
      const int r = idx4 >> 2;
      const int c = (idx4 & 3) << 2;
      *(v4f*)(&lsA[0][r * LDT + c]) = ta[i];
      *(v4f*)(&lsB[0][r * LDT + c]) = tb[i];
    }
    __syncthreads();
    mma_chunk(lsA[0], lsB[0], acc, wm, wn, lrow, khalf);
  }
#endif

  // Epilogue. C layout: VGPR r -> M = r + 8*(lane/16), N = lane%16.
#pragma unroll
  for (int i = 0; i < 2; i++) {
#pragma unroll
    for (int j = 0; j < 4; j++) {
      const int mb = m0 + wm * 32 + i * 16 + khalf * 8;
      const int nn = n0 + wn * 64 + j * 16 + lrow;
      const float bv = bias[nn];
#pragma unroll
      for (int r = 0; r < 8; r++) {
        const size_t off = (size_t)(mb + r) * N + nn;
        const float v = acc[i][j][r] + bv;
        if (GELU) {
          out[off] = 0.5f * v * (1.0f + erff(v * 0.70710678118654752f));
        } else {
          out[off] = res[off] + v;  // residual add (res may alias out)
        }
      }
    }
  }
}

// ---------------------------------------------------------------------------
extern "C" void kernel_launch(void* const* d_in, const int* in_sizes, int n_in,
                              void* d_out, int out_size, void* d_ws, size_t ws_size,
                              hipStream_t stream) {
  const float* x    = (const float*)d_in[0];
  const float* dl   = (const float*)d_in[1];
  const float* Wout = (const float*)d_in[2];
  const float* bout = (const float*)d_in[3];
  const float* cw   = (const float*)d_in[4];
  const float* cb   = (const float*)d_in[5];
  const float* g1   = (const float*)d_in[6];
  const float* bln1 = (const float*)d_in[7];
  const float* g2   = (const float*)d_in[8];
  const float* bln2 = (const float*)d_in[9];
  const float* W1   = (const float*)d_in[10];
  const float* b1   = (const float*)d_in[11];
  const float* W2   = (const float*)d_in[12];
  const float* b2   = (const float*)d_in[13];
  float* out = (float*)d_out;

  const size_t per = (size_t)B_DIM * T_DIM * D_DIM;  // 16,777,216 floats
  float* ws    = (float*)d_ws;
  float* xn    = ws;               // [B,T,D]
  float* ss    = xn + per;         // [B,T,D]  (reused as h_in after GEMM1)
  float* x1    = ss + per;         // [B,T,D]
  float* hbuf  = x1 + per;         // [B,T,HID] = 2*per
  float* carry = hbuf + 2 * per;   // [B,NCHUNK,D]
  float* hin   = ss;

  const int rows = B_DIM * T_DIM;  // 32768

  // 1) LN1
  ln_kernel<<<rows, 256, 0, stream>>>(x, xn, g1, bln1);
  // 2) parallel EMA scan
  scan_phase1<<<B_DIM * NCHUNK, 512, 0, stream>>>(xn, dl, ss, carry);
  scan_phase2<<<B_DIM, 512, 0, stream>>>(dl, carry);
  scan_phase3<<<B_DIM * NCHUNK, 512, 0, stream>>>(dl, carry, ss);
  // 3) x1 = x + depthwise conv(xn) + conv_b
  conv_res_kernel<<<(int)(per / 256), 256, 0, stream>>>(x, xn, cw, cb, x1);
  // 4) x1 += ss @ W_out^T + b_out
  gemm_f32_wmma<false><<<dim3(rows / 128, D_DIM / 128), 256, 0, stream>>>(
      ss, Wout, bout, x1, x1, rows, D_DIM, D_DIM);
  // 5) LN2 -> h_in
  ln_kernel<<<rows, 256, 0, stream>>>(x1, hin, g2, bln2);
  // 6) h = gelu(h_in @ W1^T + b1)
  gemm_f32_wmma<true><<<dim3(rows / 128, HID_DIM / 128), 256, 0, stream>>>(
      hin, W1, b1, x1, hbuf, rows, HID_DIM, D_DIM);
  // 7) out = x1 + h @ W2^T + b2
  gemm_f32_wmma<false><<<dim3(rows / 128, D_DIM / 128), 256, 0, stream>>>(
      hbuf, W2, b2, x1, out, rows, D_DIM, HID_DIM);

  (void)in_sizes; (void)n_in; (void)out_size; (void)ws_size;
}